// VectorQuantizer_43001212567560
// MI455X (gfx1250) — compile-verified
//
#include <hip/hip_runtime.h>
#include <stdint.h>

typedef __attribute__((ext_vector_type(2))) float v2f;
typedef __attribute__((ext_vector_type(4))) float v4f;
typedef __attribute__((ext_vector_type(8))) float v8f;

#define EMBED_DIM 64
#define K_SIZE 2048
#define N_TOTAL (32 * 2048)
#define WAVES_PER_WG 8
#define ROWS_PER_WAVE 16
#define ROWS_PER_WG (WAVES_PER_WG * ROWS_PER_WAVE)
#define KTILES (K_SIZE / 16)

// Precompute 0.5 * ||e_k||^2 for each codebook entry; thread 0 also zeroes the
// loss accumulator slot (stream-ordered before the main kernel).
__global__ void vq_norms_kernel(const float* __restrict__ emb,
                                float* __restrict__ norms,
                                float* __restrict__ loss_out) {
  int k = blockIdx.x * blockDim.x + threadIdx.x;
  if (k == 0) *loss_out = 0.0f;
  float s = 0.0f;
#pragma unroll
  for (int d = 0; d < EMBED_DIM; ++d) {
    float e = emb[(size_t)d * K_SIZE + k];
    s += e * e;
  }
  norms[k] = 0.5f * s;
}

__global__ __launch_bounds__(256) void vq_main_kernel(
    const float* __restrict__ x, const float* __restrict__ emb,
    const float* __restrict__ norms, float* __restrict__ outQ,
    float* __restrict__ outEnc, float* __restrict__ outIdx,
    float* __restrict__ outLoss) {
  // Double-buffered 64x16 f32 embedding tile (4 KB each).
  __shared__ float lds[2][EMBED_DIM * 16];

  const int tid = threadIdx.x;
  const int lane = tid & 31;
  const int wave = tid >> 5;
  const int hf = lane >> 4;        // which 16-lane half
  const int lcol = lane & 15;      // N column / A row within tile
  const int q0 = hf * 2;           // K sub-offset for A/B fragments
  const int rowBase = ((int)blockIdx.x * WAVES_PER_WG + wave) * ROWS_PER_WAVE;

  // ---- Preload A fragments: 16 WMMA steps of K=4 each, this wave's 16 rows.
  // A 16x4 f32 layout: lane M = lane&15; VGPR0 = K {0 | 2}, VGPR1 = K {1 | 3}.
  v2f a[16];
  {
    const float* xrow = x + (size_t)(rowBase + lcol) * EMBED_DIM;
#pragma unroll
    for (int s = 0; s < 16; ++s)
      a[s] = *(const v2f*)(xrow + 4 * s + q0);
  }

  // Cooperative async stage (memory -> LDS DMA, ASYNCcnt-tracked) of one
  // 64(d) x 16(k) f32 tile: 256 threads x 16B each.
  auto stage_async = [&](int kt, int buf) {
    int d = tid >> 2;
    int c4 = (tid & 3) * 4;
    const float* g = emb + (size_t)d * K_SIZE + kt * 16 + c4;
    unsigned ldsOff = (unsigned)(uintptr_t)&lds[buf][d * 16 + c4];
    asm volatile("global_load_async_to_lds_b128 %0, %1, off"
                 :
                 : "v"(ldsOff), "v"(g)
                 : "memory");
  };
  auto wait_async = [&]() {
    asm volatile("s_wait_asynccnt 0" ::: "memory");
  };

  stage_async(0, 0);
  wait_async();
  __syncthreads();

  float bestV[8];
  int bestI[8];
#pragma unroll
  for (int i = 0; i < 8; ++i) {
    bestV[i] = -3.4e38f;
    bestI[i] = 0;
  }

  for (int kt = 0; kt < KTILES; ++kt) {
    const int buf = kt & 1;
    if (kt + 1 < KTILES) stage_async(kt + 1, buf ^ 1);

    // Two independent accumulation chains to keep the WMMA pipe full.
    v8f c0 = {0.f, 0.f, 0.f, 0.f, 0.f, 0.f, 0.f, 0.f};
    v8f c1 = {0.f, 0.f, 0.f, 0.f, 0.f, 0.f, 0.f, 0.f};
    const float* L = &lds[buf][0];
#pragma unroll
    for (int s = 0; s < 16; s += 2) {
      // B 4x16 f32: lane N = lane&15; VGPR0 = K {0 | 2}, VGPR1 = K {1 | 3}.
      v2f b0, b1;
      b0.x = L[(4 * s + q0) * 16 + lcol];
      b0.y = L[(4 * s + q0 + 1) * 16 + lcol];
      b1.x = L[(4 * (s + 1) + q0) * 16 + lcol];
      b1.y = L[(4 * (s + 1) + q0 + 1) * 16 + lcol];
      c0 = __builtin_amdgcn_wmma_f32_16x16x4_f32(false, a[s], false, b0,
                                                 (short)0, c0, false, false);
      c1 = __builtin_amdgcn_wmma_f32_16x16x4_f32(false, a[s + 1], false, b1,
                                                 (short)0, c1, false, false);
    }

    // score = x.e - 0.5||e||^2 ; running argmax (strict > keeps first max).
    const int k = kt * 16 + lcol;
    const float nrm = norms[k];
#pragma unroll
    for (int i = 0; i < 8; ++i) {
      float sc = (c0[i] + c1[i]) - nrm;
      bool gt = sc > bestV[i];
      bestV[i] = gt ? sc : bestV[i];
      bestI[i] = gt ? k : bestI[i];
    }

    wait_async();   // our async stage of tile kt+1 is complete
    __syncthreads();  // publish to all waves; everyone done reading buf
  }

  // ---- Argmax reduction across the 16 lanes of each half (tie -> lower idx).
#pragma unroll
  for (int m = 1; m <= 8; m <<= 1) {
#pragma unroll
    for (int i = 0; i < 8; ++i) {
      float ov = __shfl_xor(bestV[i], m, 32);
      int oi = __shfl_xor(bestI[i], m, 32);
      if (ov > bestV[i] || (ov == bestV[i] && oi < bestI[i])) {
        bestV[i] = ov;
        bestI[i] = oi;
      }
    }
  }
  // Exchange halves so every lane knows all 16 row winners.
  int rem[8];
#pragma unroll
  for (int i = 0; i < 8; ++i) rem[i] = __shfl_xor(bestI[i], 16, 32);

  int rowIdx[16];
#pragma unroll
  for (int r = 0; r < 16; ++r) {
    int v0 = (r < 8) ? bestI[r] : rem[r - 8];  // view from half 0 (rows 0..7 local)
    int v1 = (r < 8) ? rem[r] : bestI[r - 8];  // view from half 1 (rows 8..15 local)
    rowIdx[r] = hf ? v1 : v0;
  }

  // ---- Indices output (stored as float per harness output dtype).
#pragma unroll
  for (int r = 0; r < 16; ++r)
    if (lane == r) outIdx[rowBase + r] = (float)rowIdx[r];

  // ---- Quantized output + commitment-loss partial.
  float lacc = 0.0f;
  const int d0 = lane * 2;
#pragma unroll
  for (int r = 0; r < 16; ++r) {
    const int bk = rowIdx[r];
    float e0 = emb[(size_t)d0 * K_SIZE + bk];
    float e1 = emb[(size_t)(d0 + 1) * K_SIZE + bk];
    size_t go = (size_t)(rowBase + r) * EMBED_DIM + d0;
    float x0 = x[go], x1 = x[go + 1];
    v2f q = {e0, e1};
    *(v2f*)(outQ + go) = q;
    float dx0 = e0 - x0, dx1 = e1 - x1;
    lacc += dx0 * dx0 + dx1 * dx1;
  }
#pragma unroll
  for (int m = 1; m <= 16; m <<= 1) lacc += __shfl_xor(lacc, m, 32);
  if (lane == 0)
    atomicAdd(outLoss, lacc * (1.0f / (float)((size_t)N_TOTAL * EMBED_DIM)));

  // ---- One-hot encodings: 512 MB stream, nontemporal 128-bit stores.
#pragma unroll 2
  for (int r = 0; r < 16; ++r) {
    const int bk = rowIdx[r];
    float* row = outEnc + (size_t)(rowBase + r) * K_SIZE;
    for (int c0i = lane * 4; c0i < K_SIZE; c0i += 128) {
      v4f v;
      v.x = (c0i + 0 == bk) ? 1.0f : 0.0f;
      v.y = (c0i + 1 == bk) ? 1.0f : 0.0f;
      v.z = (c0i + 2 == bk) ? 1.0f : 0.0f;
      v.w = (c0i + 3 == bk) ? 1.0f : 0.0f;
      __builtin_nontemporal_store(v, (v4f*)(row + c0i));
    }
  }
}

extern "C" void kernel_launch(void* const* d_in, const int* in_sizes, int n_in,
                              void* d_out, int out_size, void* d_ws,
                              size_t ws_size, hipStream_t stream) {
  const float* x = (const float*)d_in[0];     // [32, 2048, 64]
  const float* emb = (const float*)d_in[1];   // [64, 2048]
  float* out = (float*)d_out;
  float* outQ = out;                                        // 65536*64
  float* outEnc = outQ + (size_t)N_TOTAL * EMBED_DIM;       // 65536*2048
  float* outIdx = outEnc + (size_t)N_TOTAL * K_SIZE;        // 65536
  float* outLoss = outIdx + N_TOTAL;                        // 1
  float* norms = (float*)d_ws;                              // 2048 floats

  vq_norms_kernel<<<K_SIZE / 256, 256, 0, stream>>>(emb, norms, outLoss);
  vq_main_kernel<<<N_TOTAL / ROWS_PER_WG, 256, 0, stream>>>(
      x, emb, norms, outQ, outEnc, outIdx, outLoss);
}